// CategoricalEmbedding_15066745274952
// MI455X (gfx1250) — compile-verified
//
#include <hip/hip_runtime.h>

// CDNA5 / gfx1250. Wave32. Uses V_WMMA_F32_16X16X4_F32 (full-precision f32 WMMA):
// kernel is bandwidth-bound (~650MB moved vs 31.7 GFLOP), so f32 matrix math is
// free and matches the reference numerics.

typedef __attribute__((ext_vector_type(2))) float v2f;
typedef __attribute__((ext_vector_type(8))) float v8f;

#define NVARS      32
#define BATCH      8192
#define DMODEL     512
#define TILE_M     16
#define THREADS    256          // 8 waves
#define MAX_DP     228          // max d (225) padded to multiple of 4
#define OUT_STRIDE 516          // 512 + 4 pad: avoids 8-row LDS bank aliasing

struct VarPtrs {
    const float* emb[NVARS];    // emb_tables[i]: (c_i, d_i)
    const float* w[NVARS];      // proj_w[i]:     (512, d_i) row-major
};

__global__ __launch_bounds__(THREADS)
void cat_emb_ln_wmma(VarPtrs tabs,
                     const float* __restrict__ proj_b,  // (32, 512)
                     const float* __restrict__ ln_g,    // (32, 512)
                     const float* __restrict__ ln_b,    // (32, 512)
                     const int*   __restrict__ xidx,    // (8192, 32) row-major
                     float* __restrict__ out)           // (32, 8192, 512)
{
    const int var  = blockIdx.y;
    const int m0   = blockIdx.x * TILE_M;
    const int tid  = threadIdx.x;
    const int lane = tid & 31;
    const int wave = tid >> 5;

    // d_i = ceil(sqrt(c)) + 1 for c in {1000, 5000, 20000, 50000}, pattern % 4
    int d;
    switch (var & 3) {
        case 0:  d = 33;  break;
        case 1:  d = 72;  break;
        case 2:  d = 143; break;
        default: d = 225; break;
    }
    const int dp = (d + 3) & ~3;   // zero-padded K extent

    const float* __restrict__ emb = tabs.emb[var];
    const float* __restrict__ W   = tabs.w[var];

    __shared__ int   sIdx[TILE_M];
    __shared__ float sA[TILE_M * MAX_DP];          // gathered A tile (padded)
    __shared__ float sOut[TILE_M * OUT_STRIDE];    // pre-LN GEMM results

    if (tid < TILE_M) sIdx[tid] = xidx[(m0 + tid) * NVARS + var];
    __syncthreads();

    // Stage the 16 gathered embedding rows into LDS (shared A for all 8 waves).
    for (int t = tid; t < TILE_M * dp; t += THREADS) {
        const int r = t / dp;
        const int k = t - r * dp;
        sA[r * dp + k] = (k < d) ? emb[(long)sIdx[r] * d + k] : 0.0f;
    }
    __syncthreads();

    // ---------------- GEMM: each wave owns a 16M x 64N tile ----------------
    // f32 16x16x4 fragment layouts (ISA 7.12.2):
    //   A: lane L -> M=L&15 ; VGPR0=A[M][2*(L>>4)], VGPR1=A[M][2*(L>>4)+1]
    //   B: lane L -> N=L&15 ; VGPR0=B[2*(L>>4)][N], VGPR1=B[2*(L>>4)+1][N]
    //   C/D: VGPR v -> D[v + 8*(L>>4)][ (L&15) + n_base ]
    const int mA    = lane & 15;
    const int khalf = (lane >> 4) << 1;   // 0 or 2
    const int nl    = lane & 15;
    const int nb0   = wave * 64;

    v8f acc[4] = {v8f{}, v8f{}, v8f{}, v8f{}};

    const int kfull = d & ~3;
    for (int k0 = 0; k0 < kfull; k0 += 4) {
        const int kk = k0 + khalf;
        v2f a;
        a.x = sA[mA * dp + kk];
        a.y = sA[mA * dp + kk + 1];
        const float* wp = W + kk;
        const int n0 = nb0 + nl;
        v2f b0, b1, b2, b3;
        b0.x = wp[(n0     ) * d]; b0.y = wp[(n0     ) * d + 1];
        b1.x = wp[(n0 + 16) * d]; b1.y = wp[(n0 + 16) * d + 1];
        b2.x = wp[(n0 + 32) * d]; b2.y = wp[(n0 + 32) * d + 1];
        b3.x = wp[(n0 + 48) * d]; b3.y = wp[(n0 + 48) * d + 1];
        acc[0] = __builtin_amdgcn_wmma_f32_16x16x4_f32(false, a, false, b0, (short)0, acc[0], false, false);
        acc[1] = __builtin_amdgcn_wmma_f32_16x16x4_f32(false, a, false, b1, (short)0, acc[1], false, false);
        acc[2] = __builtin_amdgcn_wmma_f32_16x16x4_f32(false, a, false, b2, (short)0, acc[2], false, false);
        acc[3] = __builtin_amdgcn_wmma_f32_16x16x4_f32(false, a, false, b3, (short)0, acc[3], false, false);
    }
    if (kfull < d) {   // ragged K tail (d % 4 in {1, 3}); A is zero-padded in LDS
        const int kk = kfull + khalf;
        v2f a;
        a.x = sA[mA * dp + kk];
        a.y = sA[mA * dp + kk + 1];
        const int n0 = nb0 + nl;
        const bool v0 = (kk < d), v1 = (kk + 1 < d);
        v2f b0, b1, b2, b3;
        b0.x = v0 ? W[(n0     ) * d + kk] : 0.0f; b0.y = v1 ? W[(n0     ) * d + kk + 1] : 0.0f;
        b1.x = v0 ? W[(n0 + 16) * d + kk] : 0.0f; b1.y = v1 ? W[(n0 + 16) * d + kk + 1] : 0.0f;
        b2.x = v0 ? W[(n0 + 32) * d + kk] : 0.0f; b2.y = v1 ? W[(n0 + 32) * d + kk + 1] : 0.0f;
        b3.x = v0 ? W[(n0 + 48) * d + kk] : 0.0f; b3.y = v1 ? W[(n0 + 48) * d + kk + 1] : 0.0f;
        acc[0] = __builtin_amdgcn_wmma_f32_16x16x4_f32(false, a, false, b0, (short)0, acc[0], false, false);
        acc[1] = __builtin_amdgcn_wmma_f32_16x16x4_f32(false, a, false, b1, (short)0, acc[1], false, false);
        acc[2] = __builtin_amdgcn_wmma_f32_16x16x4_f32(false, a, false, b2, (short)0, acc[2], false, false);
        acc[3] = __builtin_amdgcn_wmma_f32_16x16x4_f32(false, a, false, b3, (short)0, acc[3], false, false);
    }

    // ------------- bias add + spill the 16x512 tile to LDS ----------------
    const int mh = (lane >> 4) << 3;   // 0 or 8
    #pragma unroll
    for (int t = 0; t < 4; ++t) {
        const int n = nb0 + 16 * t + nl;
        const float bias = proj_b[var * DMODEL + n];
        #pragma unroll
        for (int v = 0; v < 8; ++v)
            sOut[(v + mh) * OUT_STRIDE + n] = acc[t][v] + bias;
    }
    __syncthreads();

    // ---------------- LayerNorm: each wave normalizes 2 rows ---------------
    #pragma unroll
    for (int rr = 0; rr < 2; ++rr) {
        const int m = wave * 2 + rr;
        float vals[16];
        float s = 0.0f, s2 = 0.0f;
        #pragma unroll
        for (int j = 0; j < 16; ++j) {
            const float v = sOut[m * OUT_STRIDE + lane + 32 * j];
            vals[j] = v;
            s  += v;
            s2 += v * v;
        }
        #pragma unroll
        for (int off = 16; off >= 1; off >>= 1) {
            s  += __shfl_xor(s,  off, 32);
            s2 += __shfl_xor(s2, off, 32);
        }
        const float mean = s  * (1.0f / (float)DMODEL);
        const float varc = s2 * (1.0f / (float)DMODEL) - mean * mean;
        const float rstd = rsqrtf(varc + 1e-5f);
        float* __restrict__ op =
            out + ((size_t)var * BATCH + (size_t)(m0 + m)) * DMODEL;
        #pragma unroll
        for (int j = 0; j < 16; ++j) {
            const int n = lane + 32 * j;
            op[n] = (vals[j] - mean) * rstd * ln_g[var * DMODEL + n]
                    + ln_b[var * DMODEL + n];
        }
    }
}

extern "C" void kernel_launch(void* const* d_in, const int* in_sizes, int n_in,
                              void* d_out, int out_size, void* d_ws, size_t ws_size,
                              hipStream_t stream) {
    // setup_inputs order: 32 emb tables, 32 proj_w, proj_b, ln_g, ln_b, x
    VarPtrs tabs;
    for (int i = 0; i < NVARS; ++i) {
        tabs.emb[i] = (const float*)d_in[i];
        tabs.w[i]   = (const float*)d_in[NVARS + i];
    }
    const float* proj_b = (const float*)d_in[2 * NVARS + 0];
    const float* ln_g   = (const float*)d_in[2 * NVARS + 1];
    const float* ln_b   = (const float*)d_in[2 * NVARS + 2];
    const int*   x      = (const int*)  d_in[2 * NVARS + 3];

    dim3 grid(BATCH / TILE_M, NVARS);
    cat_emb_ln_wmma<<<grid, THREADS, 0, stream>>>(
        tabs, proj_b, ln_g, ln_b, x, (float*)d_out);
}